// BalancedWeightClusterLoss_82059645157780
// MI455X (gfx1250) — compile-verified
//
#include <hip/hip_runtime.h>
#include <stdint.h>

#ifndef __has_builtin
#define __has_builtin(x) 0
#endif

#define WG 256   // 8 wave32 per block

// 64KB row staging buffer + small reduction scratch, single LDS struct so
// offsets are contiguous and the row is 16B aligned for b128 traffic.
struct __align__(16) RowSmem {
  float row[16384];
  float redA[8];
  float redB[8];
  float bc[2];      // [0]=lower, [1]=step
};

__device__ __forceinline__ float wave_sum(float v) {
#pragma unroll
  for (int off = 16; off > 0; off >>= 1)
    v += __shfl_xor(v, off, 32);   // wave32 butterfly, fixed order -> deterministic
  return v;
}

__global__ __launch_bounds__(WG)
void row_cluster_loss(const float* __restrict__ W, const float* __restrict__ S,
                      float* __restrict__ partial, int K) {
  __shared__ RowSmem sm;
  const int row  = blockIdx.x;
  const int tid  = threadIdx.x;
  const int lane = tid & 31;
  const int wave = tid >> 5;

  // ---- async copy: HBM row -> LDS (CDNA5 async-to-LDS path, ASYNCcnt) ----
  const uint64_t gbase = (uint64_t)(uintptr_t)(W + (size_t)row * (size_t)K);
  // Low 32 bits of a generic pointer into LDS == LDS byte address (flat aperture).
  const uint32_t lbase = (uint32_t)(uintptr_t)(&sm.row[0]);
  const int nch = K >> 2;  // float4 chunks (K = 16384 -> 4096 chunks, 16/thread)

  for (int j = tid; j < nch; j += WG) {
    const uint32_t boff = (uint32_t)j * 16u;
    asm volatile("global_load_async_to_lds_b128 %0, %1, %2 offset:0"
                 :: "v"(lbase + boff), "v"(boff), "s"(gbase)
                 : "memory");
  }
#if __has_builtin(__builtin_amdgcn_s_wait_asynccnt)
  __builtin_amdgcn_s_wait_asynccnt(0);
  asm volatile("" ::: "memory");
#else
  asm volatile("s_wait_asynccnt 0" ::: "memory");
#endif
  __syncthreads();

  // ---- pass 1: sum and sum-of-squares from LDS ----
  float s0 = 0.0f, s1 = 0.0f;
  for (int j = tid; j < nch; j += WG) {
    const float4 v = *(const float4*)&sm.row[j * 4];
    s0 += v.x + v.y + v.z + v.w;
    s1 += v.x * v.x + v.y * v.y + v.z * v.z + v.w * v.w;
  }
  s0 = wave_sum(s0);
  s1 = wave_sum(s1);
  if (lane == 0) { sm.redA[wave] = s0; sm.redB[wave] = s1; }
  __syncthreads();
  if (tid == 0) {
    float S0 = 0.0f, S1 = 0.0f;
#pragma unroll
    for (int w = 0; w < WG / 32; ++w) { S0 += sm.redA[w]; S1 += sm.redB[w]; }
    const float fk   = (float)K;
    const float mean = S0 / fk;
    float var = (S1 - S0 * S0 / fk) / (fk - 1.0f);   // ddof=1, matches torch/jnp std
    var = fmaxf(var, 0.0f);
    const float sd = sqrtf(var);
    sm.bc[0] = mean - 2.0f * sd;          // lower = mean - STD_DEV_NUM*std
    sm.bc[1] = (4.0f * sd) / 15.0f;       // step  = 2*STD_DEV_NUM*std / NUM_Q
  }
  __syncthreads();
  const float lower = sm.bc[0];
  const float step  = sm.bc[1];
  const float sc    = S[row];             // uniform -> scalar load

  // ---- pass 2: bucket + L1 loss from LDS ----
  float loss = 0.0f;
  for (int j = tid; j < nch; j += WG) {
    const float4 v = *(const float4*)&sm.row[j * 4];
    const float xs[4] = {v.x, v.y, v.z, v.w};
#pragma unroll
    for (int e = 0; e < 4; ++e) {
      const float x = xs[e];
      float f = (x - lower) / step;                 // true divide, matches reference
      f = fminf(fmaxf(f, 0.0f), 14.0f);             // clip to [0, NUM_Q-1]
      const int qi = (int)f;                        // trunc (== floor, f >= 0)
      const float target = sc * (float)(qi - 7);    // centers[idx] = scale*(idx+THD_NEG)
      loss += fabsf(x - target);
    }
  }
  loss = wave_sum(loss);
  if (lane == 0) sm.redA[wave] = loss;
  __syncthreads();
  if (tid == 0) {
    float t = 0.0f;
#pragma unroll
    for (int w = 0; w < WG / 32; ++w) t += sm.redA[w];
    partial[row] = t;
  }
}

__global__ __launch_bounds__(WG)
void final_reduce(const float* __restrict__ partial, float* __restrict__ out, int n) {
  __shared__ float red[WG / 32];
  const int tid = threadIdx.x;
  float v = 0.0f;
  for (int i = tid; i < n; i += WG) v += partial[i];   // fixed assignment/order
  v = wave_sum(v);
  if ((tid & 31) == 0) red[tid >> 5] = v;
  __syncthreads();
  if (tid == 0) {
    float t = 0.0f;
#pragma unroll
    for (int w = 0; w < WG / 32; ++w) t += red[w];
    out[0] = t;   // COEFFICIENT == 1.0
  }
}

extern "C" void kernel_launch(void* const* d_in, const int* in_sizes, int n_in,
                              void* d_out, int out_size, void* d_ws, size_t ws_size,
                              hipStream_t stream) {
  const float* W = (const float*)d_in[0];   // [C, K] fp32
  const float* S = (const float*)d_in[1];   // [C]    fp32
  const int C = in_sizes[1];
  const int K = in_sizes[0] / C;
  float* partial = (float*)d_ws;            // C floats of scratch

  row_cluster_loss<<<C, WG, 0, stream>>>(W, S, partial, K);
  final_reduce<<<1, WG, 0, stream>>>(partial, (float*)d_out, C);
}